// SubSamplingLayer_16149077033588
// MI455X (gfx1250) — compile-verified
//
#include <hip/hip_runtime.h>
#include <hip/hip_bf16.h>

#define RES    128
#define BATCH  16
#define MPTS   (RES * RES)          // 16384 trajectory points
#define KC     32                   // K-chunk staged in LDS per iteration
#define SPLITK 4                    // deterministic split-K factor
#define KLEN   (MPTS / SPLITK)      // 4096 k-values per block
#define TWO_PI_OVER_RES 0.049087385212340517f   // 2*pi/128

typedef __attribute__((ext_vector_type(2))) float v2f;
typedef __attribute__((ext_vector_type(8))) float v8f;

// ---------------------------------------------------------------------------
// Kernel 1: bilinear sample of 2-channel k-space at trajectory points.
//   s_r/s_i : [BATCH][MPTS]
// ---------------------------------------------------------------------------
__global__ void sample_kernel(const float* __restrict__ ksp,   // [B][H][W][2]
                              const float* __restrict__ traj,  // [M][2]
                              float* __restrict__ s_r,
                              float* __restrict__ s_i) {
    int idx = blockIdx.x * blockDim.x + threadIdx.x;   // [0, BATCH*MPTS)
    int m = idx & (MPTS - 1);
    int b = idx >> 14;

    float tr = traj[2 * m + 0];
    float tc = traj[2 * m + 1];
    float pr = tr + 0.5f * RES;
    float pc = tc + 0.5f * RES;
    float r0f = floorf(pr), c0f = floorf(pc);
    float wr = pr - r0f, wc = pc - c0f;
    int r0 = min(max((int)r0f, 0), RES - 1);
    int r1 = min(r0 + 1, RES - 1);
    int c0 = min(max((int)c0f, 0), RES - 1);
    int c1 = min(c0 + 1, RES - 1);

    float w00 = (1.f - wr) * (1.f - wc);
    float w01 = (1.f - wr) * wc;
    float w10 = wr * (1.f - wc);
    float w11 = wr * wc;

    const float* base = ksp + (size_t)b * (RES * RES * 2);
    int i00 = (r0 * RES + c0) * 2;
    int i01 = (r0 * RES + c1) * 2;
    int i10 = (r1 * RES + c0) * 2;
    int i11 = (r1 * RES + c1) * 2;

    float re = w00 * base[i00]     + w01 * base[i01]
             + w10 * base[i10]     + w11 * base[i11];
    float im = w00 * base[i00 + 1] + w01 * base[i01 + 1]
             + w10 * base[i10 + 1] + w11 * base[i11 + 1];

    s_r[idx] = re;
    s_i[idx] = im;
}

// ---------------------------------------------------------------------------
// Kernel 2: DFT twiddle tables.
//   exr/exi : Ex^T stored [H][M]  (h-major, so A-fragments are contiguous in k)
//   eyr/eyi : Ey    stored [M][W]
// ---------------------------------------------------------------------------
__global__ void trig_kernel(const float* __restrict__ traj,
                            float* __restrict__ exr, float* __restrict__ exi,
                            float* __restrict__ eyr, float* __restrict__ eyi) {
    int idx = blockIdx.x * blockDim.x + threadIdx.x;   // [0, MPTS*RES)

    // Ex^T[h][m]: idx = h*MPTS + m  -> coalesced table writes + traj reads
    int m = idx & (MPTS - 1);
    int h = idx >> 14;
    float thx = TWO_PI_OVER_RES * traj[2 * m + 0] * (float)(h - RES / 2);
    float sx, cx;
    __sincosf(thx, &sx, &cx);
    exr[idx] = cx;
    exi[idx] = sx;

    // Ey[m][w]: idx = m*RES + w
    int m2 = idx >> 7;
    int w = idx & (RES - 1);
    float thy = TWO_PI_OVER_RES * traj[2 * m2 + 1] * (float)(w - RES / 2);
    float sy, cy;
    __sincosf(thy, &sy, &cy);
    eyr[idx] = cy;
    eyi[idx] = sy;
}

// ---------------------------------------------------------------------------
// Kernel 3: fused complex GEMM  out[b] = Ex^T (s[b] * Ey)  via f32 WMMA,
// split-K by SPLITK for parallelism (deterministic reduction in kernel 4).
// Grid: BATCH*8*SPLITK blocks, 256 threads (8 wave32).
//   blockIdx = (b*8 + htile)*SPLITK + ks
// Each wave owns one 16x16 complex partial tile (wtile = wave id).
// Per K-chunk: stage ExT tile + fused tmp tile in LDS, prefetch the next
// chunk's Ey tile (global_prefetch_b8), then 32x V_WMMA_F32_16X16X4_F32.
// ---------------------------------------------------------------------------
__global__ __launch_bounds__(256, 4)
void nufft_gemm_kernel(const float* __restrict__ exr,
                       const float* __restrict__ exi,
                       const float* __restrict__ eyr,
                       const float* __restrict__ eyi,
                       const float* __restrict__ s_r,
                       const float* __restrict__ s_i,
                       float* __restrict__ partial) {   // [SPLITK][B][2][H*W]
    __shared__ float sAr[16 * KC];        // ExT re tile [16 h][KC k]
    __shared__ float sAi[16 * KC];        // ExT im tile
    __shared__ float sTr[KC * RES];       // tmp re tile [KC k][128 w]
    __shared__ float sTi[KC * RES];       // tmp im tile

    const int ks    = blockIdx.x & (SPLITK - 1);
    const int htile = (blockIdx.x >> 2) & 7;
    const int b     = blockIdx.x >> 5;
    const int t     = threadIdx.x;
    const int lane  = t & 31;
    const int wtile = t >> 5;             // 0..7 -> output column tile

    const int kbeg = ks * KLEN;
    const int kend = kbeg + KLEN;

    const float* srp = s_r + (size_t)b * MPTS;
    const float* sip = s_i + (size_t)b * MPTS;

    v8f acc_rr = {}, acc_ii = {}, acc_ri = {}, acc_ir = {};

    // A-fragment addressing (32-bit A 16x4 layout, ISA 7.12.2):
    //   lanes 0-15: K = {0,1};  lanes 16-31: K = {2,3}; row M = lane & 15
    const int rowA  = lane & 15;
    const int khalf = (lane >> 4) << 1;
    const int colB  = wtile * 16 + (lane & 15);

    for (int k0 = kbeg; k0 < kend; k0 += KC) {
        __syncthreads();   // protect LDS from previous iteration's readers

        // ---- stage ExT tile: 16 x KC, re + im -------------------------------
        for (int e = t; e < 16 * KC; e += 256) {
            int hh = e >> 5;             // e / KC
            int kk = e & (KC - 1);
            int g  = (htile * 16 + hh) * MPTS + k0 + kk;
            sAr[e] = exr[g];
            sAi[e] = exi[g];
        }
        // ---- stage fused tmp tile: tmp[m][w] = s[m] * Ey[m][w] -------------
        for (int p = t; p < KC * RES; p += 256) {
            int mm = p >> 7;
            int ww = p & (RES - 1);
            int gm = k0 + mm;
            float er = eyr[gm * RES + ww];
            float ei = eyi[gm * RES + ww];
            float sr = srp[gm];
            float si = sip[gm];
            sTr[p] = sr * er - si * ei;
            sTi[p] = sr * ei + si * er;
        }
        // ---- prefetch next chunk's Ey tile while this chunk computes -------
        int knext = k0 + KC;
        if (knext < kend) {
            const float* nr = eyr + (size_t)knext * RES + t * 16;  // 64B/lane
            const float* ni = eyi + (size_t)knext * RES + t * 16;
            __builtin_prefetch(nr, 0, 1);
            __builtin_prefetch(ni, 0, 1);
        }
        __syncthreads();

        // ---- 8 k-steps x 4 products of V_WMMA_F32_16X16X4_F32 --------------
#pragma unroll
        for (int kk = 0; kk < KC; kk += 4) {
            const int ka = kk + khalf;
            v2f ar, ai, br, bi;
            ar.x = sAr[rowA * KC + ka];
            ar.y = sAr[rowA * KC + ka + 1];
            ai.x = sAi[rowA * KC + ka];
            ai.y = sAi[rowA * KC + ka + 1];
            br.x = sTr[ka * RES + colB];
            br.y = sTr[(ka + 1) * RES + colB];
            bi.x = sTi[ka * RES + colB];
            bi.y = sTi[(ka + 1) * RES + colB];

            acc_rr = __builtin_amdgcn_wmma_f32_16x16x4_f32(
                false, ar, false, br, (short)0, acc_rr, false, false);
            acc_ii = __builtin_amdgcn_wmma_f32_16x16x4_f32(
                false, ai, false, bi, (short)0, acc_ii, false, false);
            acc_ri = __builtin_amdgcn_wmma_f32_16x16x4_f32(
                false, ar, false, bi, (short)0, acc_ri, false, false);
            acc_ir = __builtin_amdgcn_wmma_f32_16x16x4_f32(
                false, ai, false, br, (short)0, acc_ir, false, false);
        }
    }

    // ---- epilogue: partial_re = rr - ii, partial_im = ri + ir --------------
    // C/D layout: VGPR r -> row r + 8*(lane>=16), col = lane & 15
    const int hbase = htile * 16 + (lane >> 4) * 8;
    const int wcol  = wtile * 16 + (lane & 15);
    float* p_re = partial + ((size_t)(ks * BATCH + b) * 2 + 0) * (RES * RES);
    float* p_im = partial + ((size_t)(ks * BATCH + b) * 2 + 1) * (RES * RES);
#pragma unroll
    for (int r = 0; r < 8; ++r) {
        float re = acc_rr[r] - acc_ii[r];
        float im = acc_ri[r] + acc_ir[r];
        int off = (hbase + r) * RES + wcol;
        p_re[off] = re;
        p_im[off] = im;
    }
}

// ---------------------------------------------------------------------------
// Kernel 4: deterministic split-K reduction (fixed order -> bitwise stable).
// ---------------------------------------------------------------------------
__global__ void reduce_kernel(const float* __restrict__ partial,
                              float* __restrict__ out) {
    int i = blockIdx.x * blockDim.x + threadIdx.x;   // [0, BATCH*2*RES*RES)
    const size_t stride = (size_t)BATCH * 2 * RES * RES;
    float a = partial[i];
    a += partial[i + 1 * stride];
    a += partial[i + 2 * stride];
    a += partial[i + 3 * stride];
    out[i] = a;
}

// ---------------------------------------------------------------------------
extern "C" void kernel_launch(void* const* d_in, const int* in_sizes, int n_in,
                              void* d_out, int out_size, void* d_ws, size_t ws_size,
                              hipStream_t stream) {
    const float* ksp  = (const float*)d_in[0];   // [16,1,128,128,2] f32
    const float* traj = (const float*)d_in[1];   // [16384,2] f32
    float* out = (float*)d_out;                  // [16,1,2,128,128] f32

    float* ws = (float*)d_ws;
    const size_t MH = (size_t)MPTS * RES;        // 2,097,152
    float* exr = ws + 0 * MH;
    float* exi = ws + 1 * MH;
    float* eyr = ws + 2 * MH;
    float* eyi = ws + 3 * MH;
    float* s_r = ws + 4 * MH;
    float* s_i = s_r + (size_t)BATCH * MPTS;
    float* partial = s_i + (size_t)BATCH * MPTS; // SPLITK*B*2*H*W floats (8MB)
    // total ws use: 4*MH + 2*B*M + SPLITK*B*2*H*W floats = ~44 MB

    sample_kernel<<<(BATCH * MPTS) / 256, 256, 0, stream>>>(ksp, traj, s_r, s_i);
    trig_kernel<<<MH / 256, 256, 0, stream>>>(traj, exr, exi, eyr, eyi);
    nufft_gemm_kernel<<<BATCH * 8 * SPLITK, 256, 0, stream>>>(
        exr, exi, eyr, eyi, s_r, s_i, partial);
    reduce_kernel<<<(BATCH * 2 * RES * RES) / 256, 256, 0, stream>>>(partial, out);
}